// RGFSConv_44521630990362
// MI455X (gfx1250) — compile-verified
//
#include <hip/hip_runtime.h>

#define CIN   256
#define COUT  512
#define HH    96
#define WW    96
#define HW    (HH*WW)          // 9216
#define BATCH 8
#define NCLS  10
#define OC    256
#define NTOT  (BATCH*HW)       // 73728
#define MASKW 384

typedef __attribute__((ext_vector_type(2))) float v2f;
typedef __attribute__((ext_vector_type(8))) float v8f;

// workspace layout in floats (≈156 MB)
#define OFF_INTER 0
#define SZ_INTER  (BATCH*COUT*HW)        // 37,748,736 floats
#define OFF_WT    (OFF_INTER + SZ_INTER)
#define SZ_WT     (9*128*COUT*2)         // 1,179,648 floats
#define OFF_SUMS  (OFF_WT + SZ_WT)
#define SZ_SUMS   (BATCH*NCLS*COUT)      // 40,960 floats
#define OFF_SEL   (OFF_SUMS + SZ_SUMS)   // ints live here
#define SZ_SEL    (BATCH*NCLS*OC)

// ---------------------------------------------------------------------------
// Repack OIHW weights -> Wt[kk][cin/2][cout][2] so the per-lane A-fragment
// load is one contiguous float2, and 16 lanes cover 128 contiguous bytes.
// Wt index: ((kk*128 + p)*512 + o)*2 + j   with cin = 2p + j
// ---------------------------------------------------------------------------
__global__ void __launch_bounds__(256) pack_weights(
    const float* __restrict__ w, float* __restrict__ wt) {
  int t = blockIdx.x * 256 + threadIdx.x;
  if (t >= SZ_WT) return;
  int j   = t & 1;
  int o   = (t >> 1) & 511;
  int p   = (t >> 10) & 127;
  int kk  = t >> 17;                 // 0..8
  int cin = 2 * p + j;
  wt[t] = w[(o * CIN + cin) * 9 + kk];
}

// ---------------------------------------------------------------------------
// Implicit-GEMM 3x3 conv with V_WMMA_F32_16X16X4_F32.
// M = Cout (512), N = B*H*W (73728), K = Cin*9 (2304).
// One block = 256 threads = 8 waves, owns a 32-wide spatial slab for ALL 512
// output channels. Wave w covers channels [64w, 64w+64) as 4 M-tiles x 2
// N-tiles = 8 accumulators, so each weight (A) fragment feeds 2 WMMAs.
// x halo tile (256ch x 3rows x 34cols, channel-paired) staged once into LDS;
// each B-fragment is one ds_load_b64 shared by 4 WMMAs.
// Per (kk,ci) step: 2 ds_load_b64 + 4 global b64 + 8 WMMA.
// ---------------------------------------------------------------------------
#define XCOLS 34
#define XS_ELEMS (128*3*XCOLS)   // 13056 float2 = 104448 B of LDS

__global__ void __launch_bounds__(256) conv_wmma(
    const float* __restrict__ x, const float* __restrict__ bias,
    const float* __restrict__ wt, float* __restrict__ inter) {
  __shared__ v2f xs[XS_ELEMS];

  const int lane = threadIdx.x & 31;
  const int wave = threadIdx.x >> 5;
  const int half = lane >> 4;           // 0: K0/K1 pair, 1: K2/K3 pair
  const int l16  = lane & 15;

  // Uniform (scalar) spatial decode: 32-wide tiles never straddle rows (96=3*32).
  const int nb  = blockIdx.x * 32;
  const int bN  = nb / HW;
  const int sp  = nb - bN * HW;
  const int oh  = sp / WW;
  const int ow0 = sp - oh * WW;         // multiple of 32

  // ---- Stage x halo tile: xs[(p*3 + rr)*34 + col] = {c=2p, c=2p+1}
  const float* xbase = x + (size_t)bN * CIN * HW;
  for (int t = threadIdx.x; t < XS_ELEMS; t += 256) {
    const int col = t % XCOLS;
    const int rr  = (t / XCOLS) % 3;
    const int p   = t / (3 * XCOLS);
    const int ih  = oh + rr - 1;
    const int iw  = ow0 + col - 1;
    v2f v = {0.0f, 0.0f};
    if (ih >= 0 && ih < HH && iw >= 0 && iw < WW) {
      const float* xp = xbase + (size_t)(2 * p) * HW + ih * WW + iw;
      v.x = xp[0];
      v.y = xp[HW];
    }
    xs[t] = v;
  }
  __syncthreads();

  // ---- Main loop
  v8f acc00 = {}, acc10 = {}, acc20 = {}, acc30 = {};   // N-tile 0
  v8f acc01 = {}, acc11 = {}, acc21 = {}, acc31 = {};   // N-tile 1
  const int mw = wave * 64;             // this wave's channel base
  const float* wl = wt + (size_t)(mw + l16) * 2;

  #pragma unroll
  for (int kk = 0; kk < 9; ++kk) {
    const int kh = kk / 3;
    const int kw = kk - 3 * kh;
    const v2f* bsrc = xs + (kh * XCOLS + l16 + kw);       // + p*(3*XCOLS)/step
    const float* wk = wl + (size_t)kk * (128 * COUT * 2); // + p*1024 per step

    #pragma unroll 2
    for (int ci = 0; ci < CIN; ci += 4) {
      const int p = (ci >> 1) + half;   // channel-pair index for this half

      const v2f bv0 = bsrc[p * (3 * XCOLS)];              // ds_load_b64
      const v2f bv1 = bsrc[p * (3 * XCOLS) + 16];         // ds_load_b64
      const float* wp = wk + (size_t)p * (COUT * 2);
      const v2f a0 = *(const v2f*)(wp);
      const v2f a1 = *(const v2f*)(wp + 16 * 2);
      const v2f a2 = *(const v2f*)(wp + 32 * 2);
      const v2f a3 = *(const v2f*)(wp + 48 * 2);

      acc00 = __builtin_amdgcn_wmma_f32_16x16x4_f32(false, a0, false, bv0,
                                                    (short)0, acc00, false, false);
      acc10 = __builtin_amdgcn_wmma_f32_16x16x4_f32(false, a1, false, bv0,
                                                    (short)0, acc10, false, false);
      acc20 = __builtin_amdgcn_wmma_f32_16x16x4_f32(false, a2, false, bv0,
                                                    (short)0, acc20, false, false);
      acc30 = __builtin_amdgcn_wmma_f32_16x16x4_f32(false, a3, false, bv0,
                                                    (short)0, acc30, false, false);
      acc01 = __builtin_amdgcn_wmma_f32_16x16x4_f32(false, a0, false, bv1,
                                                    (short)0, acc01, false, false);
      acc11 = __builtin_amdgcn_wmma_f32_16x16x4_f32(false, a1, false, bv1,
                                                    (short)0, acc11, false, false);
      acc21 = __builtin_amdgcn_wmma_f32_16x16x4_f32(false, a2, false, bv1,
                                                    (short)0, acc21, false, false);
      acc31 = __builtin_amdgcn_wmma_f32_16x16x4_f32(false, a3, false, bv1,
                                                    (short)0, acc31, false, false);
    }
  }

  // ---- Epilogue. C/D layout: column = l16 (+16 for N-tile 1),
  // row(vgpr r) = r + 8*half.
  const size_t ob0 = (size_t)bN * COUT * HW + (size_t)oh * WW + ow0 + l16;
  #pragma unroll
  for (int r = 0; r < 8; ++r) {
    const int rb = mw + r + half * 8;
    inter[ob0      + (size_t)(rb)      * HW] = acc00[r] + bias[rb];
    inter[ob0      + (size_t)(rb + 16) * HW] = acc10[r] + bias[rb + 16];
    inter[ob0      + (size_t)(rb + 32) * HW] = acc20[r] + bias[rb + 32];
    inter[ob0      + (size_t)(rb + 48) * HW] = acc30[r] + bias[rb + 48];
    inter[ob0 + 16 + (size_t)(rb)      * HW] = acc01[r] + bias[rb];
    inter[ob0 + 16 + (size_t)(rb + 16) * HW] = acc11[r] + bias[rb + 16];
    inter[ob0 + 16 + (size_t)(rb + 32) * HW] = acc21[r] + bias[rb + 32];
    inter[ob0 + 16 + (size_t)(rb + 48) * HW] = acc31[r] + bias[rb + 48];
  }
}

// ---------------------------------------------------------------------------
// Per-(b, c) masked class sums: s[b][cls][c] = sum_{h,w: m==cls} inter.
// Register-resident 10-way partial per thread, LDS float atomics to combine.
// ---------------------------------------------------------------------------
__global__ void __launch_bounds__(256) class_sums(
    const float* __restrict__ inter, const int* __restrict__ mask,
    float* __restrict__ sums) {
  const int bc = blockIdx.x;
  const int b  = bc >> 9;
  const int c  = bc & 511;
  const float* ip = inter + (size_t)(b * COUT + c) * HW;
  const int*   mp = mask + (size_t)b * MASKW * MASKW;

  float a[NCLS];
  #pragma unroll
  for (int k = 0; k < NCLS; ++k) a[k] = 0.0f;

  for (int pos = threadIdx.x; pos < HW; pos += 256) {
    const int h = pos / WW;
    const int w = pos - h * WW;
    const int cls = mp[(h * 4) * MASKW + w * 4];   // mask[:, ::4, ::4]
    const float v = ip[pos];
    #pragma unroll
    for (int k = 0; k < NCLS; ++k) a[k] += (cls == k) ? v : 0.0f;
  }

  __shared__ float csum[NCLS];
  if (threadIdx.x < NCLS) csum[threadIdx.x] = 0.0f;
  __syncthreads();
  #pragma unroll
  for (int k = 0; k < NCLS; ++k) atomicAdd(&csum[k], a[k]);   // ds_add_f32
  __syncthreads();
  if (threadIdx.x < NCLS)
    sums[(size_t)(b * NCLS + threadIdx.x) * COUT + c] = csum[threadIdx.x];
}

// ---------------------------------------------------------------------------
// Top-256 of 512 per (b, cls) with lax.top_k tie-breaking (lower index wins),
// then emit selected channel ids in ascending order (== jnp.sort(idx)).
// ---------------------------------------------------------------------------
__global__ void __launch_bounds__(512) topk_sort(
    const float* __restrict__ sums, int* __restrict__ sel) {
  const int b   = blockIdx.x / NCLS;
  const int cls = blockIdx.x - b * NCLS;
  const int t   = threadIdx.x;

  __shared__ float v[COUT];
  __shared__ int   flag[COUT];
  v[t] = sums[(size_t)(b * NCLS + cls) * COUT + t];
  __syncthreads();

  const float my = v[t];
  int rank = 0;
  for (int c2 = 0; c2 < COUT; ++c2) {
    const float o = v[c2];
    rank += (o > my) || ((o == my) && (c2 < t));
  }
  const int s = (rank < OC) ? 1 : 0;
  flag[t] = s;
  __syncthreads();

  if (s) {
    int pos = 0;
    for (int c2 = 0; c2 < t; ++c2) pos += flag[c2];
    sel[(b * NCLS + cls) * OC + pos] = t;
  }
}

// ---------------------------------------------------------------------------
// Epilogue gather: each pixel belongs to exactly one class, so
// y[b,j,h,w] = inter[b, sel[b, m[b,h,w], j], h, w]. Fully coalesced on out.
// ---------------------------------------------------------------------------
__global__ void __launch_bounds__(256) gather_out(
    const float* __restrict__ inter, const int* __restrict__ mask,
    const int* __restrict__ sel, float* __restrict__ out) {
  const size_t gid = (size_t)blockIdx.x * 256 + threadIdx.x;
  size_t t = gid;
  const int w = (int)(t % WW); t /= WW;
  const int h = (int)(t % HH); t /= HH;
  const int j = (int)(t % OC); t /= OC;
  const int b = (int)t;

  const int cls = mask[(size_t)b * MASKW * MASKW + (h * 4) * MASKW + w * 4];
  const int c   = sel[(b * NCLS + cls) * OC + j];
  out[gid] = inter[(size_t)(b * COUT + c) * HW + (size_t)h * WW + w];
}

// ---------------------------------------------------------------------------
extern "C" void kernel_launch(void* const* d_in, const int* in_sizes, int n_in,
                              void* d_out, int out_size, void* d_ws, size_t ws_size,
                              hipStream_t stream) {
  (void)in_sizes; (void)n_in; (void)out_size; (void)ws_size;
  const float* x      = (const float*)d_in[0];
  const int*   mask   = (const int*)d_in[1];
  const float* weight = (const float*)d_in[2];
  const float* bias   = (const float*)d_in[3];
  float* out = (float*)d_out;

  float* ws    = (float*)d_ws;
  float* inter = ws + OFF_INTER;
  float* wt    = ws + OFF_WT;
  float* sums  = ws + OFF_SUMS;
  int*   sel   = (int*)(ws + OFF_SEL);

  pack_weights<<<(SZ_WT + 255) / 256, 256, 0, stream>>>(weight, wt);

  conv_wmma<<<NTOT / 32, 256, 0, stream>>>(x, bias, wt, inter);  // 2304 blocks

  class_sums<<<BATCH * COUT, 256, 0, stream>>>(inter, mask, sums);
  topk_sort<<<BATCH * NCLS, COUT, 0, stream>>>(sums, sel);
  gather_out<<<(BATCH * OC * HW) / 256, 256, 0, stream>>>(inter, mask, sel, out);
}